// GraphConvolution_54580444398123
// MI455X (gfx1250) — compile-verified
//
#include <hip/hip_runtime.h>
#include <hip/hip_bf16.h>

// ---------------------------------------------------------------------------
// GraphConvolution for MI455X (gfx1250, wave32):
//   agg[n, f] = sum_{e: src[e]==n} w[e] * feat[dst[e], f]      (SpMM, sorted src)
//   out      = relu(agg @ kernel + bias)                        (WMMA f32 16x16x4)
// ---------------------------------------------------------------------------

typedef __attribute__((ext_vector_type(2))) float v2f;
typedef __attribute__((ext_vector_type(8))) float v8f;

#define FDIM 64

// ---------------------------------------------------------------------------
// Kernel 0: zero the agg accumulator (float4 grid-stride).
// ---------------------------------------------------------------------------
__global__ void gc_zero_kernel(float4* __restrict__ p, int n4) {
    int i = blockIdx.x * blockDim.x + threadIdx.x;
    const int stride = gridDim.x * blockDim.x;
    const float4 z = make_float4(0.f, 0.f, 0.f, 0.f);
    for (; i < n4; i += stride) p[i] = z;
}

// ---------------------------------------------------------------------------
// Kernel 1: segmented SpMM.
// One wave32 owns a contiguous edge range (edge_src is sorted). Each lane
// handles 2 features (2*32 = 64 = FDIM). Current segment accumulates in
// registers; flush with f32 atomic add only when the segment id changes or
// at chunk boundaries. Atomic count ~= number of segments, not edges.
// ---------------------------------------------------------------------------
__global__ void gc_spmm_kernel(const float* __restrict__ feat,
                               const float* __restrict__ ew,
                               const int*   __restrict__ esrc,
                               const int*   __restrict__ edst,
                               float*       __restrict__ agg,
                               int E) {
    const int lane   = threadIdx.x & 31;
    const int wave   = (blockIdx.x * blockDim.x + threadIdx.x) >> 5;
    const int nwaves = (gridDim.x * blockDim.x) >> 5;
    const int per    = (E + nwaves - 1) / nwaves;

    int e0 = wave * per;
    int e1 = e0 + per;
    if (e1 > E) e1 = E;
    if (e0 >= e1) return;                      // wave-uniform exit

    const int foff = lane * 2;
    float ax = 0.f, ay = 0.f;
    int cur = -1;

    for (int e = e0; e < e1; ++e) {
        const int s = esrc[e];                 // wave-uniform broadcast load
        if (s != cur) {
            if (cur >= 0) {
                float* p = agg + (size_t)cur * FDIM + foff;
                __hip_atomic_fetch_add(p,     ax, __ATOMIC_RELAXED, __HIP_MEMORY_SCOPE_AGENT);
                __hip_atomic_fetch_add(p + 1, ay, __ATOMIC_RELAXED, __HIP_MEMORY_SCOPE_AGENT);
            }
            ax = 0.f; ay = 0.f;
            cur = s;
        }
        const int   d = edst[e];
        const float w = ew[e];
        const float2 f = *reinterpret_cast<const float2*>(feat + (size_t)d * FDIM + foff);
        ax = fmaf(w, f.x, ax);
        ay = fmaf(w, f.y, ay);
    }
    if (cur >= 0) {
        float* p = agg + (size_t)cur * FDIM + foff;
        __hip_atomic_fetch_add(p,     ax, __ATOMIC_RELAXED, __HIP_MEMORY_SCOPE_AGENT);
        __hip_atomic_fetch_add(p + 1, ay, __ATOMIC_RELAXED, __HIP_MEMORY_SCOPE_AGENT);
    }
}

// ---------------------------------------------------------------------------
// Kernel 2: out = relu(agg @ W + bias) via V_WMMA_F32_16X16X4_F32.
// One wave computes one 16x16 output tile; K=64 -> 16 chained WMMAs.
// VGPR layouts per CDNA5 ISA 7.12.2:
//   A (16x4 f32):  lanes 0-15 -> M=lane, v0=K0 v1=K1; lanes 16-31 -> v0=K2 v1=K3
//   B (4x16 f32):  v0: row K0 (lanes 0-15) / K2 (lanes 16-31); v1: K1 / K3
//   C/D (16x16):   VGPR i: lanes 0-15 -> M=i, N=lane; lanes 16-31 -> M=i+8
// ---------------------------------------------------------------------------
__global__ void gc_gemm_relu_kernel(const float* __restrict__ agg,
                                    const float* __restrict__ W,
                                    const float* __restrict__ bias,
                                    float*       __restrict__ out,
                                    int totalTiles) {
    const int lane = threadIdx.x & 31;
    const int tile = blockIdx.x * (blockDim.x >> 5) + (threadIdx.x >> 5);
    if (tile >= totalTiles) return;            // wave-uniform: EXEC stays full

    const int mt   = tile >> 2;                // row tile (16 rows)
    const int nt   = tile & 3;                 // col tile (16 of 64 units)
    const int row0 = mt * 16;
    const int col0 = nt * 16;
    const int half = lane >> 4;                // 0: K(0,1)  1: K(2,3)
    const int l    = lane & 15;

    v8f acc = {};
#pragma unroll
    for (int k = 0; k < FDIM; k += 4) {
        const int ka = k + half * 2;
        const float2 av = *reinterpret_cast<const float2*>(
            agg + (size_t)(row0 + l) * FDIM + ka);
        v2f a; a.x = av.x; a.y = av.y;

        const float* bp = W + (size_t)ka * FDIM + col0 + l;
        v2f b; b.x = bp[0]; b.y = bp[FDIM];

        acc = __builtin_amdgcn_wmma_f32_16x16x4_f32(
            /*neg_a=*/false, a, /*neg_b=*/false, b,
            /*c_mod=*/(short)0, acc, /*reuse_a=*/false, /*reuse_b=*/false);
    }

    const float bv = bias[col0 + l];
    float* op = out + (size_t)(row0 + half * 8) * FDIM + col0 + l;
#pragma unroll
    for (int i = 0; i < 8; ++i) {
        const float v = acc[i] + bv;
        op[(size_t)i * FDIM] = v > 0.f ? v : 0.f;
    }
}

// ---------------------------------------------------------------------------
// Launch
// ---------------------------------------------------------------------------
extern "C" void kernel_launch(void* const* d_in, const int* in_sizes, int n_in,
                              void* d_out, int out_size, void* d_ws, size_t ws_size,
                              hipStream_t stream) {
    const float* feat = (const float*)d_in[0];   // [N, 64]
    const float* ew   = (const float*)d_in[1];   // [E]
    const float* W    = (const float*)d_in[2];   // [64, 64]
    const float* bias = (const float*)d_in[3];   // [64]
    const int*   esrc = (const int*)  d_in[4];   // [E] sorted
    const int*   edst = (const int*)  d_in[5];   // [E]
    float*       out  = (float*)d_out;           // [N, 64]
    float*       agg  = (float*)d_ws;            // [N, 64] scratch

    const int N = in_sizes[0] / FDIM;
    const int E = in_sizes[4];

    // 1) agg = 0
    const int n4 = N * (FDIM / 4);
    gc_zero_kernel<<<512, 256, 0, stream>>>((float4*)agg, n4);

    // 2) segmented SpMM: 512 blocks x 8 waves -> 4096 waves, ~196 edges/wave
    gc_spmm_kernel<<<512, 256, 0, stream>>>(feat, ew, esrc, edst, agg, E);

    // 3) WMMA GEMM + bias + ReLU: 1 wave per 16x16 tile, 4 waves per block
    const int tilesM     = (N + 15) / 16;        // N=50000 -> 3125 exact
    const int totalTiles = tilesM * 4;           // 12500
    const int grid       = (totalTiles + 3) / 4;
    gc_gemm_relu_kernel<<<grid, 128, 0, stream>>>(agg, W, bias, out, totalTiles);
}